// ModulatedDeformableConv2d_82119774699851
// MI455X (gfx1250) — compile-verified
//
#include <hip/hip_runtime.h>
#include <hip/hip_bf16.h>
#include <math.h>

// ---------------------------------------------------------------------------
// Modulated Deformable Conv2d (DCNv2) forward for gfx1250 (MI455X).
//   B=4, C=64, H=W=128, O=64, K=3, stride=1, pad=1, dil=1, mask_scale=2
// Fused implicit-im2col GEMM on fp32 matrix cores (V_WMMA_F32_16X16X4_F32):
// the problem is L2-gather bound (~600M 4B gathers, ~7 GFLOP), so fp32 WMMA
// keeps exact reference precision at zero cost.  Double-buffered K-chunks
// overlap the bilinear gathers with the WMMA stream; weight chunks are staged
// with GLOBAL_LOAD_ASYNC_TO_LDS_B128 (ASYNCcnt-tracked) when available.
// ---------------------------------------------------------------------------

typedef float v2f __attribute__((ext_vector_type(2)));
typedef float v8f __attribute__((ext_vector_type(8)));

#define B_   4
#define C_   64
#define H_   128
#define W_   128
#define O_   64
#define KK_  9
#define P_   (H_ * W_)        // 16384 output pixels
#define KTOT (C_ * KK_)       // 576 = GEMM K
#define NB   32               // pixels (N columns) per block
#define KC   16               // K chunk (4 wmma steps of K=4)
#define NCHUNK (KTOT / KC)    // 36
#define SWSTR 20              // sW row stride: 80B -> 16B-aligned B128 dsts

#if __has_builtin(__builtin_amdgcn_global_load_async_to_lds_b128) && \
    __has_builtin(__builtin_amdgcn_s_wait_asynccnt)
#define USE_ASYNC_LDS 1
typedef int v4i_ __attribute__((vector_size(16)));   // builtin's pointee type
#else
#define USE_ASYNC_LDS 0
#endif

// ---------------------------------------------------------------------------
// Kernel 1: 27-channel offset/mask conv.  One thread per (b, pixel) computes
// ALL 27 output channels so the 3x3x64 input patch is read exactly once
// (9x less x traffic than one-thread-per-tap).  w_off indices are wave-
// uniform -> scalar constant-cache loads.  Writes py/px/mask to workspace.
// ---------------------------------------------------------------------------
__global__ __launch_bounds__(256)
void dcn_offset_kernel(const float* __restrict__ x,
                       const float* __restrict__ w_off,
                       float* __restrict__ pyA,
                       float* __restrict__ pxA,
                       float* __restrict__ mA)
{
    int gid = blockIdx.x * blockDim.x + threadIdx.x;   // (b, pixel)
    if (gid >= B_ * P_) return;
    int b = gid >> 14;          // / P_
    int p = gid & (P_ - 1);
    int oy = p >> 7;
    int ox = p & 127;

    float acc[27];
    #pragma unroll
    for (int t = 0; t < 27; ++t) acc[t] = 0.f;

    const float* xb = x + (size_t)b * C_ * P_;
    for (int c = 0; c < C_; ++c) {
        const float* xc = xb + c * P_;
        #pragma unroll
        for (int ky = 0; ky < 3; ++ky) {
            int iy = oy - 1 + ky;
            if (iy < 0 || iy >= H_) continue;
            #pragma unroll
            for (int kx = 0; kx < 3; ++kx) {
                int ix = ox - 1 + kx;
                if (ix < 0 || ix >= W_) continue;
                float v  = xc[iy * W_ + ix];
                int   wi = c * KK_ + ky * 3 + kx;        // uniform -> s_load
                #pragma unroll
                for (int t = 0; t < 27; ++t)
                    acc[t] = fmaf(v, w_off[t * KTOT + wi], acc[t]);
            }
        }
    }

    #pragma unroll
    for (int kk = 0; kk < KK_; ++kk) {
        float py = (float)(oy - 1 + (kk / 3)) + acc[kk];        // base+tap+o1
        float px = (float)(ox - 1 + (kk % 3)) + acc[9 + kk];    // base+tap+o2
        float mv = 2.0f / (1.0f + expf(-acc[18 + kk]));         // sigmoid*2
        int o = (b * KK_ + kk) * P_ + p;                        // coalesced
        pyA[o] = py;  pxA[o] = px;  mA[o] = mv;
    }
}

// ---------------------------------------------------------------------------
// Kernel 2: fused bilinear-sample + modulate + GEMM (+bias).
// Block = 32 pixels x all 64 outputs; 8 waves = 4 Mtiles x 2 Ntiles.
// Double-buffered K-chunks of 16: stage W[64][16] (async B128 -> LDS) and
// S[16][32] (4 gathers/elem) for chunk i+1 while running chunk i's WMMAs.
// One barrier per chunk.
//
// WMMA fp32 fragment layouts (ISA 7.12.2):
//   A 16x4 : a[v]   = A[m = lane%16][k = v + 2*(lane/16)]
//   B  4x16: b[v]   = B[k = v + 2*(lane/16)][n = lane%16]
//   C/D    : acc[v] = D[m = v + 8*(lane/16)][n = lane%16]
// ---------------------------------------------------------------------------
__global__ __launch_bounds__(256)
void dcn_main_kernel(const float* __restrict__ x,
                     const float* __restrict__ weight,
                     const float* __restrict__ bias,
                     const float* __restrict__ pyA,
                     const float* __restrict__ pxA,
                     const float* __restrict__ mA,
                     float* __restrict__ out)
{
    __shared__ int   cIdx[KK_][NB][4];      // clamped gather indices
    __shared__ float cWt [KK_][NB][4];      // bilinear*valid*mask weights
    __shared__ float sS[2][KC][NB + 1];     // sampled chunk, stride 33
    __shared__ float sW[2][O_][SWSTR];      // weight chunk, stride 20

    const int tid   = threadIdx.x;
    const int b     = blockIdx.y;
    const int p0    = blockIdx.x * NB;
    const int lane  = tid & 31;
    const int wave  = tid >> 5;
    const int mtile = wave >> 1;
    const int ntile = wave & 1;

    // ---- per-block coordinate table: 9 taps x 32 pixels (288 entries) ----
    for (int i = tid; i < KK_ * NB; i += 256) {
        int kk = i / NB, n = i - kk * NB;
        int base = (b * KK_ + kk) * P_ + p0 + n;
        float py = pyA[base], px = pxA[base], mv = mA[base];
        float y0 = floorf(py), x0 = floorf(px);
        float ly = py - y0,    lx = px - x0;
        float wy0 = 1.f - ly,  wx0 = 1.f - lx;
        #pragma unroll
        for (int t = 0; t < 4; ++t) {
            int dy = t >> 1, dx = t & 1;
            float cy = y0 + (float)dy, cx = x0 + (float)dx;
            bool valid = (cy >= 0.f) && (cy <= (float)(H_ - 1)) &&
                         (cx >= 0.f) && (cx <= (float)(W_ - 1));
            int yi = (int)fminf(fmaxf(cy, 0.f), (float)(H_ - 1));
            int xi = (int)fminf(fmaxf(cx, 0.f), (float)(W_ - 1));
            cIdx[kk][n][t] = yi * W_ + xi;
            float wbl = (dy ? ly : wy0) * (dx ? lx : wx0);
            cWt[kk][n][t] = valid ? wbl * mv : 0.f;
        }
    }
    __syncthreads();   // table visible before any fill reads it

    // ---- chunk stage: weights (async B128 -> LDS) + fused im2col samples ---
    auto fill_chunk = [&](int kc, int buf) {
#if USE_ASYNC_LDS
        {   // 256 threads x one 16B quad = W[64][16]; dst stride 80B (aligned)
            int m = tid >> 2, q = tid & 3;
            __builtin_amdgcn_global_load_async_to_lds_b128(
                (__attribute__((address_space(1))) v4i_*)
                    (weight + m * KTOT + kc + q * 4),
                (__attribute__((address_space(3))) v4i_*)&sW[buf][m][q * 4],
                0, 0);
        }
#else
        #pragma unroll
        for (int it = 0; it < (O_ * KC) / 256; ++it) {
            int m = it * 16 + (tid >> 4), kl = tid & 15;
            sW[buf][m][kl] = weight[m * KTOT + kc + kl];
        }
#endif
        #pragma unroll
        for (int it = 0; it < KC / 8; ++it) {       // 16x32 elems, 2 per thread
            int kl = it * 8 + wave;                  // row = wave
            int n  = lane;                           // col = lane (coalesced)
            int k  = kc + kl;
            int c  = k / KK_;
            int kk = k - c * KK_;
            const float* xb = x + (size_t)(b * C_ + c) * P_;
            const int*   id = cIdx[kk][n];
            const float* wv = cWt[kk][n];
            sS[buf][kl][n] = wv[0] * xb[id[0]] + wv[1] * xb[id[1]]
                           + wv[2] * xb[id[2]] + wv[3] * xb[id[3]];
        }
    };

    v8f acc = {};
    fill_chunk(0, 0);   // prologue

    for (int i = 0; i < NCHUNK; ++i) {
        int cur = i & 1;
#if USE_ASYNC_LDS
        __builtin_amdgcn_s_wait_asynccnt(0);   // async LDS writes done (ours)
#endif
        __syncthreads();                       // chunk i visible to all waves
        if (i + 1 < NCHUNK)
            fill_chunk((i + 1) * KC, cur ^ 1); // overlap gathers with WMMA

        #pragma unroll
        for (int s = 0; s < 4; ++s) {
            int kA = s * 4 + 2 * (lane >> 4);
            v2f a, bf;
            a.x  = sW[cur][mtile * 16 + (lane & 15)][kA];
            a.y  = sW[cur][mtile * 16 + (lane & 15)][kA + 1];
            bf.x = sS[cur][kA    ][ntile * 16 + (lane & 15)];
            bf.y = sS[cur][kA + 1][ntile * 16 + (lane & 15)];
            acc = __builtin_amdgcn_wmma_f32_16x16x4_f32(
                      false, a, false, bf, (short)0, acc, false, false);
        }
    }

    // ---- epilogue: bias + store (coalesced across lanes) ----
    int n     = lane & 15;
    int col   = p0 + ntile * 16 + n;
    int half8 = (lane >> 4) * 8;
    #pragma unroll
    for (int v = 0; v < 8; ++v) {
        int m = mtile * 16 + half8 + v;
        out[(size_t)(b * O_ + m) * P_ + col] = acc[v] + bias[m];
    }
}

// ---------------------------------------------------------------------------
extern "C" void kernel_launch(void* const* d_in, const int* in_sizes, int n_in,
                              void* d_out, int out_size, void* d_ws, size_t ws_size,
                              hipStream_t stream)
{
    const float* x      = (const float*)d_in[0];
    const float* weight = (const float*)d_in[1];
    const float* bias   = (const float*)d_in[2];
    const float* w_off  = (const float*)d_in[3];
    float* out = (float*)d_out;

    const size_t N1 = (size_t)B_ * KK_ * P_;   // 589,824 coords (7.1 MB ws)
    float* pyA = (float*)d_ws;
    float* pxA = pyA + N1;
    float* mA  = pxA + N1;

    int total1 = B_ * P_;                       // one thread per (b, pixel)
    dcn_offset_kernel<<<(total1 + 255) / 256, 256, 0, stream>>>(x, w_off, pyA, pxA, mA);

    dim3 grid2(P_ / NB, B_);                    // 512 pixel-tiles x 4 batches
    dcn_main_kernel<<<grid2, 256, 0, stream>>>(x, weight, bias, pyA, pxA, mA, out);
}